// Attention_56212531970690
// MI455X (gfx1250) — compile-verified
//
#include <hip/hip_runtime.h>
#include <hip/hip_bf16.h>
#include <math.h>

typedef __attribute__((ext_vector_type(2))) float v2f;
typedef __attribute__((ext_vector_type(8))) float v8f;

#define BATCH 16
#define DIMC  192
#define HH    84
#define WWID  84
#define HWSZ  (HH*WWID)          // 7056
#define SCALE_F 0.20412414523193154f   // (192/8)^-0.5 = 1/sqrt(24)

__device__ __forceinline__ float gelu_f(float x) {
    return 0.5f * x * (1.0f + erff(x * 0.70710678118654752f));
}

// ---------------------------------------------------------------------------
// conv1x1 as WMMA GEMM:  Y[b,o,p] = sum_k W[o,k] * (gelu?)(X[b,k,p]) + bias[o]
// 256 threads = 8 wave32; 16-pixel x 192-channel tile staged in LDS, shared
// across 8 o-tiles; each wave does 48 x v_wmma_f32_16x16x4_f32.
// Fragment layouts per CDNA5 ISA 7.12.2:
//   A (16x4):  lane l -> m=l&15, k-pair base = (l>>4)*2, 2 VGPRs
//   B (4x16):  lane l -> n=l&15, k-pair base = (l>>4)*2, 2 VGPRs
//   C/D 16x16: lane l -> n=l&15, VGPR v -> M = v + 8*(l>>4)
// ---------------------------------------------------------------------------
__global__ __launch_bounds__(256)
void conv1x1_wmma_kernel(const float* __restrict__ X, const float* __restrict__ W,
                         const float* __restrict__ bias, float* __restrict__ Y,
                         int Cout, int apply_gelu_in)
{
    __shared__ float sX[DIMC * 16];   // [k][n], 12 KB
    const int bb = blockIdx.y;
    const int p0 = blockIdx.x * 16;
    const size_t xbase = (size_t)bb * DIMC * HWSZ;

    for (int idx = threadIdx.x; idx < DIMC * 16; idx += 256) {
        int k = idx >> 4, n = idx & 15;
        float v = X[xbase + (size_t)k * HWSZ + p0 + n];
        if (apply_gelu_in) v = gelu_f(v);
        sX[idx] = v;
    }
    __syncthreads();

    const int wave = threadIdx.x >> 5;
    const int lane = threadIdx.x & 31;
    const int o0 = (blockIdx.z * 8 + wave) * 16;
    if (o0 >= Cout) return;              // wave-uniform: EXEC stays all-ones

    const int ln   = lane & 15;          // m for A, n for B/D
    const int half = lane >> 4;
    const int kb   = half * 2;

    const float* wrow = W + (size_t)(o0 + ln) * DIMC;
    __builtin_prefetch(wrow, 0, 3);

    v8f acc = {};
    for (int kk = 0; kk < DIMC; kk += 4) {
        if ((kk & 31) == 0) __builtin_prefetch(wrow + kk + 32, 0, 1);
        v2f a, b;
        a.x = wrow[kk + kb];
        a.y = wrow[kk + kb + 1];
        b.x = sX[(kk + kb) * 16 + ln];
        b.y = sX[(kk + kb + 1) * 16 + ln];
        acc = __builtin_amdgcn_wmma_f32_16x16x4_f32(
                  false, a, false, b, (short)0, acc, false, false);
    }

    const size_t ybase = (size_t)bb * Cout * HWSZ;
    for (int v = 0; v < 8; ++v) {
        int mo = v + 8 * half;
        Y[ybase + (size_t)(o0 + mo) * HWSZ + p0 + ln] = acc[v] + bias[o0 + mo];
    }
}

// ---------------------------------------------------------------------------
// Pooled-K precompute:
//   kp_ss [b][48][12][12] : 7x7 mean of qkv channels 144..191
//   kp_hor[b][48][12]     : mean over 7-row band x all cols of ch 240..287
//   kp_ver[b][48][12]     : mean over all rows x 7-col band of ch 336..383
// ---------------------------------------------------------------------------
__global__ void pool_kernel(const float* __restrict__ qkv,
                            float* __restrict__ kp_ss,
                            float* __restrict__ kp_hor,
                            float* __restrict__ kp_ver)
{
    int idx = blockIdx.x * 256 + threadIdx.x;
    const int N1 = BATCH * 48 * 144;
    const int N2 = BATCH * 48 * 12;
    if (idx < N1) {
        int b = idx / (48 * 144); int r = idx % (48 * 144);
        int c = r / 144; int kk = r % 144; int kh = kk / 12, kw = kk % 12;
        const float* src = qkv + ((size_t)b * 384 + 144 + c) * HWSZ;
        float s = 0.f;
        for (int dy = 0; dy < 7; ++dy)
            for (int dx = 0; dx < 7; ++dx)
                s += src[(kh * 7 + dy) * WWID + kw * 7 + dx];
        kp_ss[idx] = s * (1.0f / 49.0f);
    } else if (idx < N1 + N2) {
        int j = idx - N1;
        int b = j / (48 * 12); int r = j % (48 * 12); int c = r / 12; int row = r % 12;
        const float* src = qkv + ((size_t)b * 384 + 240 + c) * HWSZ;
        float s = 0.f;
        for (int dy = 0; dy < 7; ++dy)
            for (int xx = 0; xx < WWID; ++xx)
                s += src[(row * 7 + dy) * WWID + xx];
        kp_hor[j] = s * (1.0f / 588.0f);
    } else if (idx < N1 + 2 * N2) {
        int j = idx - N1 - N2;
        int b = j / (48 * 12); int r = j % (48 * 12); int c = r / 12; int col = r % 12;
        const float* src = qkv + ((size_t)b * 384 + 336 + c) * HWSZ;
        float s = 0.f;
        for (int yy = 0; yy < HH; ++yy)
            for (int dx = 0; dx < 7; ++dx)
                s += src[yy * WWID + col * 7 + dx];
        kp_ver[j] = s * (1.0f / 588.0f);
    }
}

// ---------------------------------------------------------------------------
// Windowed SSMSA: one workgroup per (batch, 7x7 window), 2 heads, hd=24.
// q = qkv ch 0..47, k = v = qkv ch 48..95.  Writes mid ch 0..47.
// ---------------------------------------------------------------------------
__global__ __launch_bounds__(128)
void win_ssmsa_kernel(const float* __restrict__ qkv,
                      const float* __restrict__ lw_dw, const float* __restrict__ lb_dw,
                      const float* __restrict__ lw_pw, const float* __restrict__ lb_pw,
                      float* __restrict__ mid)
{
    __shared__ float sQ[2][49][24];
    __shared__ float sK[2][49][24];
    __shared__ float sAH[2][7][7][7];   // [n][i][j][x]  row-attn
    __shared__ float sAW[2][7][7][7];   // [n][j][i][i2] col-attn
    __shared__ float sA[2][49][49];     // attention weights [n][q][k]
    __shared__ float sT[49][2][49];     // fusion dwconv+gelu temp [q][n][k]

    const int b  = blockIdx.y;
    const int hs = blockIdx.x / 12, wsp = blockIdx.x % 12;
    const int py0 = hs * 7, px0 = wsp * 7;

    for (int idx = threadIdx.x; idx < 2352; idx += 128) {
        int n = idx / 1176; int r = idx % 1176; int pix = r / 24; int d = r % 24;
        int i = pix / 7, j = pix % 7;
        size_t gp = (size_t)(py0 + i) * WWID + px0 + j;
        sQ[n][pix][d] = qkv[((size_t)b * 384 + n * 24 + d) * HWSZ + gp] * SCALE_F;
        sK[n][pix][d] = qkv[((size_t)b * 384 + 48 + n * 24 + d) * HWSZ + gp];
    }
    __syncthreads();

    // row-softmax (aH) and column-softmax (aW)
    for (int r = threadIdx.x; r < 196; r += 128) {
        if (r < 98) {
            int n = r / 49; int rem = r % 49; int i = rem / 7; int j = rem % 7;
            float lg[7]; float mx = -1e30f;
            for (int x = 0; x < 7; ++x) {
                float s = 0.f;
                for (int d = 0; d < 24; ++d) s += sQ[n][i*7+j][d] * sK[n][i*7+x][d];
                lg[x] = s; mx = fmaxf(mx, s);
            }
            float sum = 0.f;
            for (int x = 0; x < 7; ++x) { lg[x] = expf(lg[x] - mx); sum += lg[x]; }
            float inv = 1.0f / sum;
            for (int x = 0; x < 7; ++x) sAH[n][i][j][x] = lg[x] * inv;
        } else {
            int rr = r - 98; int n = rr / 49; int rem = rr % 49;
            int j = rem / 7; int i = rem % 7;
            float lg[7]; float mx = -1e30f;
            for (int i2 = 0; i2 < 7; ++i2) {
                float s = 0.f;
                for (int d = 0; d < 24; ++d) s += sQ[n][i*7+j][d] * sK[n][i2*7+j][d];
                lg[i2] = s; mx = fmaxf(mx, s);
            }
            float sum = 0.f;
            for (int i2 = 0; i2 < 7; ++i2) { lg[i2] = expf(lg[i2] - mx); sum += lg[i2]; }
            float inv = 1.0f / sum;
            for (int i2 = 0; i2 < 7; ++i2) sAW[n][j][i][i2] = lg[i2] * inv;
        }
    }
    __syncthreads();

    // atten[n, q=(i,j), k=(i2,x)] = aH[n,i,j,x] * aW[n, col=x, i, i2]
    for (int idx = threadIdx.x; idx < 4802; idx += 128) {
        int n = idx / 2401; int r = idx % 2401; int qq = r / 49; int kk = r % 49;
        int i = qq / 7, j = qq % 7, i2 = kk / 7, x = kk % 7;
        sA[n][qq][kk] = sAH[n][i][j][x] * sAW[n][x][i][i2];
    }
    __syncthreads();

    // fusion branch: depthwise 3x3 (SAME) + bias + gelu on each query's 2x7x7 map
    for (int idx = threadIdx.x; idx < 4802; idx += 128) {
        int n = idx / 2401; int r = idx % 2401; int qq = r / 49; int kk = r % 49;
        int i2 = kk / 7, x = kk % 7;
        float s = 0.f;
        for (int dy = 0; dy < 3; ++dy) {
            int iy = i2 + dy - 1; if (iy < 0 || iy >= 7) continue;
            for (int dx = 0; dx < 3; ++dx) {
                int ix = x + dx - 1; if (ix < 0 || ix >= 7) continue;
                s += sA[n][qq][iy * 7 + ix] * lw_dw[n * 9 + dy * 3 + dx];
            }
        }
        sT[qq][n][kk] = gelu_f(s + lb_dw[n]);
    }
    __syncthreads();

    // pointwise 2x2 mix + residual add into weights
    for (int idx = threadIdx.x; idx < 4802; idx += 128) {
        int n = idx / 2401; int r = idx % 2401; int qq = r / 49; int kk = r % 49;
        sA[n][qq][kk] += lw_pw[n * 2 + 0] * sT[qq][0][kk]
                       + lw_pw[n * 2 + 1] * sT[qq][1][kk] + lb_pw[n];
    }
    __syncthreads();

    // av = A @ V  (V == K), write mid channels 0..47
    for (int idx = threadIdx.x; idx < 2352; idx += 128) {
        int n = idx / 1176; int r = idx % 1176; int qq = r / 24; int d = r % 24;
        float s = 0.f;
        for (int kk = 0; kk < 49; ++kk) s += sA[n][qq][kk] * sK[n][kk][d];
        int i = qq / 7, j = qq % 7;
        size_t gp = (size_t)(py0 + i) * WWID + px0 + j;
        mid[((size_t)b * 192 + n * 24 + d) * HWSZ + gp] = s;
    }
}

// ---------------------------------------------------------------------------
// Pooled-grid SSMSA: q = qkv ch 96..143, K pooled (kp_ss). 16 queries/block,
// 1 wave (thread = head x query). Writes mid ch 48..95.
// ---------------------------------------------------------------------------
__global__ __launch_bounds__(32)
void gp_ssmsa_kernel(const float* __restrict__ qkv, const float* __restrict__ kp_ss,
                     const float* __restrict__ gw_dw, const float* __restrict__ gb_dw,
                     const float* __restrict__ gw_pw, const float* __restrict__ gb_pw,
                     float* __restrict__ mid)
{
    __shared__ float sK5[2][24][144];   // pooled K, both heads
    __shared__ float sL [16][2][144];   // logits -> combined weights
    __shared__ float sTt[16][2][144];   // fusion temp

    const int b = blockIdx.y;
    for (int idx = threadIdx.x; idx < 6912; idx += 32) {
        int c = idx / 144, kk = idx % 144;
        sK5[c / 24][c % 24][kk] = kp_ss[((size_t)b * 48 + c) * 144 + kk];
    }
    __syncthreads();

    const int n  = threadIdx.x >> 4;
    const int pi = threadIdx.x & 15;
    const int p  = blockIdx.x * 16 + pi;     // 441*16 == 7056 exactly

    float q[24];
    const float* qb = qkv + ((size_t)b * 384 + 96 + n * 24) * HWSZ + p;
    for (int d = 0; d < 24; ++d) q[d] = qb[(size_t)d * HWSZ] * SCALE_F;

    float* L = &sL[pi][n][0];
    for (int kk = 0; kk < 144; ++kk) {
        float s = 0.f;
        for (int d = 0; d < 24; ++d) s += q[d] * sK5[n][d][kk];
        L[kk] = s;
    }

    // combined weight = rowSoftmax(L) * colSoftmax(L)
    float rmax[12], rsum[12], cmax[12], csum[12];
    for (int r = 0; r < 12; ++r) {
        float m = -1e30f;
        for (int c = 0; c < 12; ++c) m = fmaxf(m, L[r * 12 + c]);
        float s = 0.f;
        for (int c = 0; c < 12; ++c) s += expf(L[r * 12 + c] - m);
        rmax[r] = m; rsum[r] = s;
    }
    for (int c = 0; c < 12; ++c) {
        float m = -1e30f;
        for (int r = 0; r < 12; ++r) m = fmaxf(m, L[r * 12 + c]);
        float s = 0.f;
        for (int r = 0; r < 12; ++r) s += expf(L[r * 12 + c] - m);
        cmax[c] = m; csum[c] = s;
    }
    for (int r = 0; r < 12; ++r)
        for (int c = 0; c < 12; ++c) {
            float l = L[r * 12 + c];
            L[r * 12 + c] = (expf(l - rmax[r]) / rsum[r]) * (expf(l - cmax[c]) / csum[c]);
        }

    // fusion: depthwise 3x3 + gelu on 12x12 map
    float* T = &sTt[pi][n][0];
    for (int yy = 0; yy < 12; ++yy)
        for (int xx = 0; xx < 12; ++xx) {
            float s = 0.f;
            for (int dy = 0; dy < 3; ++dy) {
                int iy = yy + dy - 1; if (iy < 0 || iy >= 12) continue;
                for (int dx = 0; dx < 3; ++dx) {
                    int ix = xx + dx - 1; if (ix < 0 || ix >= 12) continue;
                    s += L[iy * 12 + ix] * gw_dw[n * 9 + dy * 3 + dx];
                }
            }
            T[yy * 12 + xx] = gelu_f(s + gb_dw[n]);
        }
    __syncthreads();   // pointwise mixes both heads' T

    for (int kk = 0; kk < 144; ++kk)
        L[kk] += gw_pw[n * 2 + 0] * sTt[pi][0][kk]
               + gw_pw[n * 2 + 1] * sTt[pi][1][kk] + gb_pw[n];

    float* ob = mid + ((size_t)b * 192 + 48 + n * 24) * HWSZ + p;
    for (int d = 0; d < 24; ++d) {
        float s = 0.f;
        for (int kk = 0; kk < 144; ++kk) s += L[kk] * sK5[n][d][kk];
        ob[(size_t)d * HWSZ] = s;
    }
}

// ---------------------------------------------------------------------------
// Pooled MSA (horizontal / vertical): 12 keys, values == pooled keys.
// ---------------------------------------------------------------------------
__global__ __launch_bounds__(256)
void gp_msa_kernel(const float* __restrict__ qkv, const float* __restrict__ kp,
                   int qoff, float* __restrict__ mid, int midoff)
{
    __shared__ float sKP[2][24][12];
    const int b = blockIdx.y;
    for (int idx = threadIdx.x; idx < 576; idx += 256) {
        int c = idx / 12, k = idx % 12;
        sKP[c / 24][c % 24][k] = kp[((size_t)b * 48 + c) * 12 + k];
    }
    __syncthreads();

    const int n  = threadIdx.x >> 7;
    const int pi = threadIdx.x & 127;
    const int p  = blockIdx.x * 128 + pi;
    if (p >= HWSZ) return;

    float q[24];
    const float* qb = qkv + ((size_t)b * 384 + qoff + n * 24) * HWSZ + p;
    for (int d = 0; d < 24; ++d) q[d] = qb[(size_t)d * HWSZ] * SCALE_F;

    float lg[12]; float mx = -1e30f;
    for (int k = 0; k < 12; ++k) {
        float s = 0.f;
        for (int d = 0; d < 24; ++d) s += q[d] * sKP[n][d][k];
        lg[k] = s; mx = fmaxf(mx, s);
    }
    float sum = 0.f;
    for (int k = 0; k < 12; ++k) { lg[k] = expf(lg[k] - mx); sum += lg[k]; }
    float inv = 1.0f / sum;

    float* ob = mid + ((size_t)b * 192 + midoff + n * 24) * HWSZ + p;
    for (int d = 0; d < 24; ++d) {
        float s = 0.f;
        for (int k = 0; k < 12; ++k) s += lg[k] * sKP[n][d][k];
        ob[(size_t)d * HWSZ] = s * inv;
    }
}

// ---------------------------------------------------------------------------
// Host-side launcher.  Workspace needs ~261 MB:
//   qkv  : 16*384*7056 f32   (173.4 MB)
//   mid  : 16*192*7056 f32   ( 86.7 MB)
//   kp_ss/kp_hor/kp_ver      (  0.5 MB)
// ---------------------------------------------------------------------------
extern "C" void kernel_launch(void* const* d_in, const int* in_sizes, int n_in,
                              void* d_out, int out_size, void* d_ws, size_t ws_size,
                              hipStream_t stream)
{
    const float* x      = (const float*)d_in[0];
    const float* w_qkv  = (const float*)d_in[1];
    const float* b_qkv  = (const float*)d_in[2];
    const float* lw_dw  = (const float*)d_in[3];
    const float* lb_dw  = (const float*)d_in[4];
    const float* lw_pw  = (const float*)d_in[5];
    const float* lb_pw  = (const float*)d_in[6];
    const float* gw_dw  = (const float*)d_in[7];
    const float* gb_dw  = (const float*)d_in[8];
    const float* gw_pw  = (const float*)d_in[9];
    const float* gb_pw  = (const float*)d_in[10];
    const float* w_proj = (const float*)d_in[11];
    const float* b_proj = (const float*)d_in[12];
    float* out = (float*)d_out;

    float* qkv    = (float*)d_ws;
    float* mid    = qkv + (size_t)BATCH * 2 * DIMC * HWSZ;
    float* kp_ss  = mid + (size_t)BATCH * DIMC * HWSZ;
    float* kp_hor = kp_ss + BATCH * 48 * 144;
    float* kp_ver = kp_hor + BATCH * 48 * 12;

    // 1) QKV projection (WMMA GEMM): 384 x 192 x 112896
    conv1x1_wmma_kernel<<<dim3(HWSZ / 16, BATCH, 3), 256, 0, stream>>>(
        x, w_qkv, b_qkv, qkv, 2 * DIMC, 0);

    // 2) pooled keys for the three global branches
    pool_kernel<<<(BATCH * 48 * (144 + 12 + 12) + 255) / 256, 256, 0, stream>>>(
        qkv, kp_ss, kp_hor, kp_ver);

    // 3) four attention branches -> mid channels [0..47 | 48..95 | 96..143 | 144..191]
    win_ssmsa_kernel<<<dim3(144, BATCH), 128, 0, stream>>>(
        qkv, lw_dw, lb_dw, lw_pw, lb_pw, mid);
    gp_ssmsa_kernel<<<dim3(HWSZ / 16, BATCH), 32, 0, stream>>>(
        qkv, kp_ss, gw_dw, gb_dw, gw_pw, gb_pw, mid);
    gp_msa_kernel<<<dim3((HWSZ + 127) / 128, BATCH), 256, 0, stream>>>(
        qkv, kp_hor, 192, mid, 96);
    gp_msa_kernel<<<dim3((HWSZ + 127) / 128, BATCH), 256, 0, stream>>>(
        qkv, kp_ver, 288, mid, 144);

    // 4) gelu + output projection (WMMA GEMM): 192 x 192 x 112896
    conv1x1_wmma_kernel<<<dim3(HWSZ / 16, BATCH, 2), 256, 0, stream>>>(
        mid, w_proj, b_proj, out, DIMC, 1);
}